// NeighborSearch_39530878992386
// MI455X (gfx1250) — compile-verified
//
#include <hip/hip_runtime.h>

// ---------------------------------------------------------------------------
// Fixed-radius neighbor search, MI455X (gfx1250, wave32).
//   sq[m][n] = |q_m|^2 + |d_n|^2 - 2 * (Q . D^T)[m][n]
// - Gram tiles via V_WMMA_F32_16X16X4_F32 (K=3 zero-padded to 4; f32 WMMA is
//   required: f16 input rounding (~6e-3 in sq) would swamp r^2 = 0.0064).
// - The full data array (16384 x 3 f32 = 192 KB) is staged into LDS once per
//   workgroup with a single Tensor Data Mover DMA (tensor_load_to_lds),
//   waited with s_wait_tensorcnt, then B-tiles are fed from LDS.
// - Count kernel: per-lane hit accumulators + one end-of-loop shfl_xor
//   butterfly (no per-tile cross-lane ops).
// - Write kernel: wave32 ballot + prefix popcount for ordered compaction.
// ---------------------------------------------------------------------------

#define NUM_DATA     16384
#define NUM_QUERIES  16384
#define MAX_NBRS     64
#define MAX_TOTAL    (NUM_QUERIES * MAX_NBRS)
#define DATA_ELEMS   (NUM_DATA * 3)          // 49152 f32 = 192 KB in LDS

typedef float        v2f __attribute__((ext_vector_type(2)));
typedef float        v8f __attribute__((ext_vector_type(8)));
typedef unsigned int v4u __attribute__((ext_vector_type(4)));
typedef int          v8i __attribute__((ext_vector_type(8)));
typedef int          v4i __attribute__((ext_vector_type(4)));

__device__ __forceinline__ unsigned ballot32(bool p) {
  return __builtin_amdgcn_ballot_w32(p);
}

// ---------------------------------------------------------------------------
// One-shot TDM DMA: stage `nelem` f32 (1-D tensor, 4-byte elements) from
// global memory into workgroup LDS at offset 0.  D# layout per ISA ch.8:
//   group0: [1:0]count=1 | [63:32]lds_addr=0 | [120:64]global_addr |
//           [127:126]type=2
//   group1: [17:16]data_size=2(4B) | [79:48]tensor_dim0 | [111:80]tensor_dim1=1
//           | [127:112]tile_dim0 | [143:128]tile_dim1=0 (1-D tile)
//           | [207:160]tensor_dim0_stride
// This toolchain (clang-23 / therock-10.0) declares the 6-arg builtin:
//   (uint32x4 g0, int32x8 g1, int32x4 g2, int32x4 g3, int32x8 extra, i32 cpol)
// Groups 2/3 and the extra operand are unused for a <=2-D tensor -> zeros.
// Issue from one wave only; EXEC is ignored by tensor ops.
// ---------------------------------------------------------------------------
__device__ __forceinline__ void tdm_stage_f32(const float* gsrc,
                                              unsigned nelem) {
  const unsigned long long ga = (unsigned long long)(const void*)gsrc;
  v4u g0 = { 1u,                                     // count=1 (user mode)
             0u,                                     // lds_addr = 0
             (unsigned)(ga & 0xffffffffull),         // global_addr[31:0]
             (unsigned)((ga >> 32) & 0x01ffffffull)  // global_addr[56:32]
                 | 0x80000000u };                    // type=2 (bits 127:126)
  v8i g1 = { (int)(2u << 16),                        // data_size = 4 bytes
             (int)((nelem & 0xffffu) << 16),         // tensor_dim0[15:0]
             (int)(((nelem >> 16) & 0xffffu)         // tensor_dim0[31:16]
                 | (1u << 16)),                      // tensor_dim1 = 1
             (int)((nelem & 0xffffu) << 16),         // tile_dim0 (16-bit)
             0,                                      // tile_dim1/2 = 0 (1-D)
             (int)nelem,                             // tensor_dim0_stride
             0, 0 };
  v4i g2 = { 0, 0, 0, 0 };
  v4i g3 = { 0, 0, 0, 0 };
  v8i g4 = { 0, 0, 0, 0, 0, 0, 0, 0 };               // unused extra operand
  __builtin_amdgcn_tensor_load_to_lds(g0, g1, g2, g3, g4, 0);
}

// ---------------------------------------------------------------------------
// Pass 1: per-query neighbor counts.
// Wave owns a 16-query strip.  WMMA operand layout (ISA 7.12.2):
//   A (16x4): lanes 0-15 {K0,K1} for M=lane; lanes 16-31 {K2,K3=0}.
//   B (4x16): lanes 0-15 {K0,K1} for N=lane; lanes 16-31 {K2,K3=0}.
//   C/D: VGPR r -> row r (lanes 0-15) / row r+8 (lanes 16-31), N = lane%16.
// ---------------------------------------------------------------------------
__global__ void nbr_count_kernel(const float* __restrict__ data,
                                 const float* __restrict__ queries,
                                 const float* __restrict__ radius,
                                 int* __restrict__ counts) {
  extern __shared__ float lds_data[];   // DATA_ELEMS floats, TDM-staged

  if (threadIdx.x < 32) {               // wave 0 issues the DMA
    tdm_stage_f32(data, DATA_ELEMS);
    __builtin_amdgcn_s_wait_tensorcnt(0);
  }
  __syncthreads();

  const int lane  = threadIdx.x & 31;
  const int wave  = threadIdx.x >> 5;
  const int strip = blockIdx.x * (blockDim.x >> 5) + wave;
  const int m0    = strip * 16;
  const bool lo   = lane < 16;
  const int  l16  = lane & 15;

  v2f a;
  {
    const int m = m0 + l16;
    if (lo) { a.x = queries[m * 3 + 0]; a.y = queries[m * 3 + 1]; }
    else    { a.x = queries[m * 3 + 2]; a.y = 0.0f; }
  }

  float q2v[8], rad2v[8];
#pragma unroll
  for (int r = 0; r < 8; ++r) {
    const int m = m0 + r + (lo ? 0 : 8);
    const float qx = queries[m * 3 + 0];
    const float qy = queries[m * 3 + 1];
    const float qz = queries[m * 3 + 2];
    q2v[r] = qx * qx + qy * qy + qz * qz;
    const float rr = radius[m];
    rad2v[r] = rr * rr;   // dist<=r  <=>  sq<=r^2 (sqrt monotone, clamp moot)
  }

  // Per-lane hit counters: cnt8[r] counts hits of (row r|r+8, col lane%16).
  int cnt8[8] = {0, 0, 0, 0, 0, 0, 0, 0};

  for (int n0 = 0; n0 < NUM_DATA; n0 += 16) {
    const int n = n0 + l16;
    const float dx = lds_data[n * 3 + 0];
    const float dy = lds_data[n * 3 + 1];
    const float dz = lds_data[n * 3 + 2];
    const float d2 = dx * dx + dy * dy + dz * dz;

    v2f b;
    b.x = lo ? dx : dz;
    b.y = lo ? dy : 0.0f;

    v8f c = {};
    v8f dm = __builtin_amdgcn_wmma_f32_16x16x4_f32(
        false, a, false, b, (short)0, c, false, false);

#pragma unroll
    for (int r = 0; r < 8; ++r) {
      const float sq = q2v[r] + d2 - 2.0f * dm[r];
      cnt8[r] += (sq <= rad2v[r]);    // v_cmp + add, no cross-lane work
    }
  }

  // Butterfly-reduce over the 16 lanes of each half (masks 1,2,4,8 stay
  // within a half).  Afterwards every low lane holds totals for rows 0..7,
  // every high lane for rows 8..15.
#pragma unroll
  for (int msk = 1; msk <= 8; msk <<= 1) {
#pragma unroll
    for (int r = 0; r < 8; ++r) cnt8[r] += __shfl_xor(cnt8[r], msk, 32);
  }

  if ((lane & 15) == 0) {
    const int base = m0 + (lo ? 0 : 8);
#pragma unroll
    for (int r = 0; r < 8; ++r) counts[base + r] = cnt8[r];
  }
}

// ---------------------------------------------------------------------------
// Pass 2: exclusive scan of 16384 counts -> splits[0..16384] (single block).
// ---------------------------------------------------------------------------
__global__ void nbr_scan_kernel(const int* __restrict__ counts,
                                int* __restrict__ splits) {
  __shared__ int s[1024];
  const int tid = threadIdx.x;

  int loc[16];
  int sum = 0;
#pragma unroll
  for (int j = 0; j < 16; ++j) {
    loc[j] = sum;
    sum += counts[tid * 16 + j];
  }
  s[tid] = sum;
  __syncthreads();

  for (int off = 1; off < 1024; off <<= 1) {
    const int v = (tid >= off) ? s[tid - off] : 0;
    __syncthreads();
    s[tid] += v;
    __syncthreads();
  }

  const int base = (tid == 0) ? 0 : s[tid - 1];
#pragma unroll
  for (int j = 0; j < 16; ++j) splits[tid * 16 + j] = base + loc[j];
  if (tid == 1023) splits[NUM_QUERIES] = s[1023];
}

// ---------------------------------------------------------------------------
// Pass 3: sentinel fill of neighbors_index.
// ---------------------------------------------------------------------------
__global__ void nbr_fill_kernel(int* __restrict__ out_idx) {
  const int i = blockIdx.x * blockDim.x + threadIdx.x;
  if (i < MAX_TOTAL) out_idx[i] = NUM_DATA;
}

// ---------------------------------------------------------------------------
// Pass 4: recompute masks (same FP ops -> identical results) and write packed
// column indices at splits[row] + rank (row-major, ascending columns).
// ---------------------------------------------------------------------------
__global__ void nbr_write_kernel(const float* __restrict__ data,
                                 const float* __restrict__ queries,
                                 const float* __restrict__ radius,
                                 const int* __restrict__ splits,
                                 int* __restrict__ out_idx) {
  extern __shared__ float lds_data[];

  if (threadIdx.x < 32) {
    tdm_stage_f32(data, DATA_ELEMS);
    __builtin_amdgcn_s_wait_tensorcnt(0);
  }
  __syncthreads();

  const int lane  = threadIdx.x & 31;
  const int wave  = threadIdx.x >> 5;
  const int strip = blockIdx.x * (blockDim.x >> 5) + wave;
  const int m0    = strip * 16;
  const bool lo   = lane < 16;
  const int  l16  = lane & 15;

  v2f a;
  {
    const int m = m0 + l16;
    if (lo) { a.x = queries[m * 3 + 0]; a.y = queries[m * 3 + 1]; }
    else    { a.x = queries[m * 3 + 2]; a.y = 0.0f; }
  }

  float q2v[8], rad2v[8];
  int base[8];
#pragma unroll
  for (int r = 0; r < 8; ++r) {
    const int m = m0 + r + (lo ? 0 : 8);
    const float qx = queries[m * 3 + 0];
    const float qy = queries[m * 3 + 1];
    const float qz = queries[m * 3 + 2];
    q2v[r] = qx * qx + qy * qy + qz * qz;
    const float rr = radius[m];
    rad2v[r] = rr * rr;
    base[r] = splits[m];
  }

  for (int n0 = 0; n0 < NUM_DATA; n0 += 16) {
    const int n = n0 + l16;
    const float dx = lds_data[n * 3 + 0];
    const float dy = lds_data[n * 3 + 1];
    const float dz = lds_data[n * 3 + 2];
    const float d2 = dx * dx + dy * dy + dz * dz;

    v2f b;
    b.x = lo ? dx : dz;
    b.y = lo ? dy : 0.0f;

    v8f c = {};
    v8f dm = __builtin_amdgcn_wmma_f32_16x16x4_f32(
        false, a, false, b, (short)0, c, false, false);

#pragma unroll
    for (int r = 0; r < 8; ++r) {
      const float sq  = q2v[r] + d2 - 2.0f * dm[r];
      const bool  inr = sq <= rad2v[r];
      const unsigned bal = ballot32(inr);
      if (inr) {
        const int rank = lo
            ? __popc(bal & ((1u << lane) - 1u))                 // row r
            : __popc((bal >> 16) & ((1u << (lane - 16)) - 1u)); // row r+8
        const int idx = base[r] + rank;
        if (idx < MAX_TOTAL) out_idx[idx] = n;
      }
      base[r] += lo ? __popc(bal & 0xffffu) : __popc(bal >> 16);
    }
  }
}

// ---------------------------------------------------------------------------
extern "C" void kernel_launch(void* const* d_in, const int* in_sizes, int n_in,
                              void* d_out, int out_size, void* d_ws,
                              size_t ws_size, hipStream_t stream) {
  (void)in_sizes; (void)n_in; (void)out_size; (void)ws_size;

  const float* data    = (const float*)d_in[0];
  const float* queries = (const float*)d_in[1];
  const float* radius  = (const float*)d_in[2];

  int* out_idx = (int*)d_out;              // [MAX_TOTAL] neighbors_index
  int* splits  = out_idx + MAX_TOTAL;      // [NUM_QUERIES+1]
  int* counts  = (int*)d_ws;               // [NUM_QUERIES] scratch (64 KB)

  const int strips = NUM_QUERIES / 16;     // 1024 waves of work
  const dim3 blk(256);                     // 8 waves / block (wave32)
  const dim3 grd(strips / 8);              // 128 blocks
  const size_t lds_bytes = DATA_ELEMS * sizeof(float);  // 192 KB / WGP

  nbr_count_kernel<<<grd, blk, lds_bytes, stream>>>(data, queries, radius,
                                                    counts);
  nbr_scan_kernel<<<1, 1024, 0, stream>>>(counts, splits);
  nbr_fill_kernel<<<MAX_TOTAL / 256, 256, 0, stream>>>(out_idx);
  nbr_write_kernel<<<grd, blk, lds_bytes, stream>>>(data, queries, radius,
                                                    splits, out_idx);
}